// CompGCN_dg_mtg_60988535603571
// MI455X (gfx1250) — compile-verified
//
#include <hip/hip_runtime.h>

typedef __attribute__((ext_vector_type(2))) float v2f;
typedef __attribute__((ext_vector_type(8))) float v8f;

#define DF 128

// ---------------------------------------------------------------------------
// D(16x16) += A(16x4) * B(4x16), fp32 WMMA.
// A frag: lanes 0-15 row M=lane {K=ka,ka+1}; lanes 16-31 {K=ka+2,ka+3} (ka=k0+2*half)
// B frag (weights stored TRANSPOSED, WT[j][k]): b = WT[col][ka..ka+1] -> one b64 load
// ---------------------------------------------------------------------------
__device__ __forceinline__ v8f wmma_step(v2f a, v2f b, v8f c) {
  return __builtin_amdgcn_wmma_f32_16x16x4_f32(false, a, false, b, (short)0, c,
                                               false, false);
}

// ---------------------------------------------------------------------------
__global__ void k_zero(float* __restrict__ p, size_t n) {
  size_t i = (size_t)blockIdx.x * blockDim.x + threadIdx.x;
  size_t stride = (size_t)gridDim.x * blockDim.x;
  for (; i < n; i += stride) p[i] = 0.0f;
}

// Prep: transpose weights for b64 B-fragment loads, fold W_text into W_rel's
// bottom half (linearity of the per-edge path), fuse biases.
//   WT_text[j*128+k] = W_text[k][j]
//   WT_inv [j*256+k] = W_inv[k][j]                       (k = 0..255)
//   WT_edge[j*256+k] = k<128 ? W_rel[k][j]
//                            : sum_m W_text[k-128][m] * W_rel[128+m][j]
//   b_fus[j] = b_rel[j] + sum_m b_text[m] * W_rel[128+m][j]
__global__ void k_prep(const float* __restrict__ W_text, const float* __restrict__ b_text,
                       const float* __restrict__ W_inv, const float* __restrict__ W_rel,
                       const float* __restrict__ b_rel,
                       float* __restrict__ WT_text, float* __restrict__ WT_inv,
                       float* __restrict__ WT_edge, float* __restrict__ b_fus) {
  const int t = blockIdx.x * blockDim.x + threadIdx.x;   // 0 .. 128*256-1
  if (t >= DF * 2 * DF) return;
  const int j = t >> 8;          // output column 0..127
  const int k = t & 255;         // K index 0..255
  WT_inv[(size_t)j * 256 + k] = W_inv[(size_t)k * DF + j];
  if (k < DF) {
    WT_text[(size_t)j * DF + k] = W_text[(size_t)k * DF + j];
    WT_edge[(size_t)j * 256 + k] = W_rel[(size_t)k * DF + j];
  } else {
    const int kk = k - DF;
    float acc = 0.0f;
    for (int m = 0; m < DF; ++m)
      acc = fmaf(W_text[(size_t)kk * DF + m], W_rel[(size_t)(DF + m) * DF + j], acc);
    WT_edge[(size_t)j * 256 + k] = acc;
  }
  if (t < DF) {
    float acc = b_rel[t];
    for (int m = 0; m < DF; ++m)
      acc = fmaf(b_text[m], W_rel[(size_t)(DF + m) * DF + t], acc);
    b_fus[t] = acc;
  }
}

// Wave-per-edge: acc_s[dst] += s_h[e]; acc_r[dst] += h[dst]*e_h[e]; deg[dst] += 1
__global__ void k_edge_scatter(const float* __restrict__ h, const float* __restrict__ e_h,
                               const float* __restrict__ s_h, const int* __restrict__ dst,
                               float* __restrict__ acc_s, float* __restrict__ acc_r,
                               float* __restrict__ degf, int E) {
  const int e    = (int)(((size_t)blockIdx.x * blockDim.x + threadIdx.x) >> 5);
  const int lane = threadIdx.x & 31;
  if (e >= E) return;
  const int d = dst[e];
  const float4 sh = ((const float4*)(s_h + (size_t)e * DF))[lane];
  const float4 eh = ((const float4*)(e_h + (size_t)e * DF))[lane];
  const float4 hv = ((const float4*)(h + (size_t)d * DF))[lane];
  float* as = acc_s + (size_t)d * DF + lane * 4;
  float* ar = acc_r + (size_t)d * DF + lane * 4;
  atomicAdd(as + 0, sh.x); atomicAdd(as + 1, sh.y);
  atomicAdd(as + 2, sh.z); atomicAdd(as + 3, sh.w);
  atomicAdd(ar + 0, eh.x * hv.x); atomicAdd(ar + 1, eh.y * hv.y);
  atomicAdd(ar + 2, eh.z * hv.z); atomicAdd(ar + 3, eh.w * hv.w);
  if (lane == 0) atomicAdd(degf + d, 1.0f);
}

// Node-side double GEMM (one wave = 16 nodes):
//   h_o_s = (acc_s/denom) @ W_text + (deg/denom)*b_text
//   hsro  = [h_o_s | acc_r/denom] @ W_inv + b_inv
__global__ __launch_bounds__(64) void k_node_gemm(
    const float* __restrict__ acc_s, const float* __restrict__ acc_r,
    const float* __restrict__ degf,
    const float* __restrict__ WT_text, const float* __restrict__ b_text,
    const float* __restrict__ WT_inv, const float* __restrict__ b_inv,
    float* __restrict__ hsro, int N) {
  __shared__ float sA[2][16][DF];   // acc_s/denom
  __shared__ float sR[2][16][DF];   // acc_r/denom
  __shared__ float sS[2][16][DF];   // h_o_s
  __shared__ float sScale[2][16];   // deg/denom
  const int wave = threadIdx.x >> 5;
  const int lane = threadIdx.x & 31;
  const int half = lane >> 4;
  const int l16  = lane & 15;
  const int r0   = (blockIdx.x * 2 + wave) * 16;
  if (r0 >= N) return;               // wave-uniform: EXEC stays all-ones

  for (int r = 0; r < 16; ++r) {
    int node = r0 + r; if (node >= N) node = N - 1;
    const float dg  = degf[node];
    const float inv = 1.0f / fmaxf(dg, 1.0f);
    if (lane == 0) sScale[wave][r] = dg * inv;
    const float4 a  = ((const float4*)(acc_s + (size_t)node * DF))[lane];
    const float4 rr = ((const float4*)(acc_r + (size_t)node * DF))[lane];
    float4 va; va.x = a.x * inv;  va.y = a.y * inv;  va.z = a.z * inv;  va.w = a.w * inv;
    float4 vr; vr.x = rr.x * inv; vr.y = rr.y * inv; vr.z = rr.z * inv; vr.w = rr.w * inv;
    ((float4*)sA[wave][r])[lane] = va;
    ((float4*)sR[wave][r])[lane] = vr;
  }

  // GEMM1: sS = sA @ W_text (+ scaled bias)    [B frags: b64 from WT_text]
  for (int n0 = 0; n0 < DF; n0 += 16) {
    const size_t bcol = (size_t)(n0 + l16) * DF;
    v8f c = {};
    for (int k0 = 0; k0 < DF; k0 += 4) {
      const int ka = k0 + 2 * half;
      const v2f a = *(const v2f*)&sA[wave][l16][ka];
      const v2f b = *(const v2f*)(WT_text + bcol + ka);
      c = wmma_step(a, b, c);
    }
    const float bt = b_text[n0 + l16];
#pragma unroll
    for (int i = 0; i < 8; ++i) {
      const int row = i + 8 * half;
      sS[wave][row][n0 + l16] = c[i] + bt * sScale[wave][row];
    }
  }

  // GEMM2: hsro = [sS | sR] @ W_inv + b_inv  (K = 256, b64 B frags from WT_inv)
  for (int n0 = 0; n0 < DF; n0 += 16) {
    const size_t bcol = (size_t)(n0 + l16) * 256;
    const float bi = b_inv[n0 + l16];
    v8f c;
#pragma unroll
    for (int i = 0; i < 8; ++i) c[i] = bi;
    for (int k0 = 0; k0 < DF; k0 += 4) {
      const int ka = k0 + 2 * half;
      const v2f a = *(const v2f*)&sS[wave][l16][ka];
      const v2f b = *(const v2f*)(WT_inv + bcol + ka);
      c = wmma_step(a, b, c);
    }
    for (int k0 = 0; k0 < DF; k0 += 4) {
      const int ka = k0 + 2 * half;
      const v2f a = *(const v2f*)&sR[wave][l16][ka];
      const v2f b = *(const v2f*)(WT_inv + bcol + DF + ka);
      c = wmma_step(a, b, c);
    }
#pragma unroll
    for (int i = 0; i < 8; ++i) {
      const int row = r0 + i + 8 * half;
      if (row < N) hsro[(size_t)row * DF + n0 + l16] = c[i];
    }
  }
}

// Wave-per-edge: out_h[dst] += hsro[src]
__global__ void k_hnew_scatter(const float* __restrict__ hsro, const int* __restrict__ src,
                               const int* __restrict__ dst, float* __restrict__ hacc, int E) {
  const int e    = (int)(((size_t)blockIdx.x * blockDim.x + threadIdx.x) >> 5);
  const int lane = threadIdx.x & 31;
  if (e >= E) return;
  const int s = src[e], d = dst[e];
  const float4 v = ((const float4*)(hsro + (size_t)s * DF))[lane];
  float* p = hacc + (size_t)d * DF + lane * 4;
  atomicAdd(p + 0, v.x); atomicAdd(p + 1, v.y);
  atomicAdd(p + 2, v.z); atomicAdd(p + 3, v.w);
}

// h_new = h_acc * norm + bias_v   (in place on d_out)
__global__ void k_hfinal(float* __restrict__ hout, const float* __restrict__ norm,
                         const float* __restrict__ bias_v, int N) {
  const int n    = (int)(((size_t)blockIdx.x * blockDim.x + threadIdx.x) >> 5);
  const int lane = threadIdx.x & 31;
  if (n >= N) return;
  const float nv = norm[n];
  const float4 b = ((const float4*)bias_v)[lane];
  float4* p = (float4*)(hout + (size_t)n * DF) + lane;
  float4 v = *p;
  v.x = fmaf(v.x, nv, b.x); v.y = fmaf(v.y, nv, b.y);
  v.z = fmaf(v.z, nv, b.z); v.w = fmaf(v.w, nv, b.w);
  *p = v;
}

// Edge GEMM, one wave = 32 edges (two 16-row tiles share every B fragment):
//   e_out = [e_h | s_h] @ WT_edge^T + b_fus      (K = 256)
// Phase 1 streams e_h through the LDS tile, phase 2 re-stages it with s_h,
// all 8 n-tile accumulator pairs stay live (128 VGPRs). Output bounces
// through the same LDS tile for fully coalesced b128 stores.
__global__ __launch_bounds__(64) void k_edge_out(
    const float* __restrict__ e_h, const float* __restrict__ s_h,
    const float* __restrict__ WT_edge, const float* __restrict__ b_fus,
    float* __restrict__ e_out, int E) {
  __shared__ float sBuf[2][32][DF];          // 32 KB total
  const int wave = threadIdx.x >> 5;
  const int lane = threadIdx.x & 31;
  const int half = lane >> 4;
  const int l16  = lane & 15;
  const long e0  = ((long)blockIdx.x * 2 + wave) * 32;
  if (e0 >= E) return;                       // wave-uniform

  // init accumulators with fused bias
  v8f acc0[8], acc1[8];
#pragma unroll
  for (int n = 0; n < 8; ++n) {
    const float bf = b_fus[n * 16 + l16];
#pragma unroll
    for (int i = 0; i < 8; ++i) { acc0[n][i] = bf; acc1[n][i] = bf; }
  }

  // ---- phase 1: A = e_h rows, K = 0..127 of WT_edge ----
  for (int r = 0; r < 32; ++r) {
    long e = e0 + r; if (e >= E) e = E - 1;
    ((float4*)sBuf[wave][r])[lane] = ((const float4*)(e_h + (size_t)e * DF))[lane];
  }
  for (int k0 = 0; k0 < DF; k0 += 4) {
    const int ka = k0 + 2 * half;
    const v2f a0 = *(const v2f*)&sBuf[wave][l16][ka];
    const v2f a1 = *(const v2f*)&sBuf[wave][16 + l16][ka];
#pragma unroll
    for (int n = 0; n < 8; ++n) {
      const v2f b = *(const v2f*)(WT_edge + (size_t)(n * 16 + l16) * 256 + ka);
      acc0[n] = wmma_step(a0, b, acc0[n]);
      acc1[n] = wmma_step(a1, b, acc1[n]);
    }
  }

  // ---- phase 2: A = s_h rows, K = 128..255 of WT_edge ----
  for (int r = 0; r < 32; ++r) {
    long e = e0 + r; if (e >= E) e = E - 1;
    ((float4*)sBuf[wave][r])[lane] = ((const float4*)(s_h + (size_t)e * DF))[lane];
  }
  for (int k0 = 0; k0 < DF; k0 += 4) {
    const int ka = k0 + 2 * half;
    const v2f a0 = *(const v2f*)&sBuf[wave][l16][ka];
    const v2f a1 = *(const v2f*)&sBuf[wave][16 + l16][ka];
#pragma unroll
    for (int n = 0; n < 8; ++n) {
      const v2f b = *(const v2f*)(WT_edge + (size_t)(n * 16 + l16) * 256 + DF + ka);
      acc0[n] = wmma_step(a0, b, acc0[n]);
      acc1[n] = wmma_step(a1, b, acc1[n]);
    }
  }

  // ---- transpose through LDS, then coalesced b128 stores ----
#pragma unroll
  for (int n = 0; n < 8; ++n) {
#pragma unroll
    for (int i = 0; i < 8; ++i) {
      sBuf[wave][i + 8 * half][n * 16 + l16]      = acc0[n][i];
      sBuf[wave][16 + i + 8 * half][n * 16 + l16] = acc1[n][i];
    }
  }
  for (int r = 0; r < 32; ++r) {
    const long e = e0 + r;
    if (e < E)
      ((float4*)(e_out + (size_t)e * DF))[lane] = ((const float4*)sBuf[wave][r])[lane];
  }
}

// ---------------------------------------------------------------------------
extern "C" void kernel_launch(void* const* d_in, const int* in_sizes, int n_in,
                              void* d_out, int out_size, void* d_ws, size_t ws_size,
                              hipStream_t stream) {
  const float* h      = (const float*)d_in[0];
  const float* e_h    = (const float*)d_in[1];
  const float* s_h    = (const float*)d_in[2];
  const float* norm   = (const float*)d_in[3];
  const int*   src    = (const int*)d_in[4];
  const int*   dst    = (const int*)d_in[5];
  const float* W_text = (const float*)d_in[6];
  const float* b_text = (const float*)d_in[7];
  const float* W_inv  = (const float*)d_in[8];
  const float* b_inv  = (const float*)d_in[9];
  const float* W_rel  = (const float*)d_in[10];
  const float* b_rel  = (const float*)d_in[11];
  const float* bias_v = (const float*)d_in[12];

  const int N = in_sizes[0] / DF;
  const int E = in_sizes[1] / DF;

  float* out_h = (float*)d_out;               // [N,128]
  float* out_e = out_h + (size_t)N * DF;      // [E,128]

  // workspace layout
  float* acc_s   = (float*)d_ws;              // N*128
  float* acc_r   = acc_s + (size_t)N * DF;    // N*128
  float* degf    = acc_r + (size_t)N * DF;    // N (padded)
  const size_t npad = ((size_t)N + 63) & ~(size_t)63;
  float* hsro    = degf + npad;               // N*128
  float* WT_text = hsro + (size_t)N * DF;     // 128*128
  float* WT_inv  = WT_text + DF * DF;         // 128*256
  float* WT_edge = WT_inv + DF * 256;         // 128*256
  float* b_fus   = WT_edge + DF * 256;        // 128

  // 1) zero accumulators (acc_s, acc_r, deg contiguous) and the h output
  k_zero<<<2048, 256, 0, stream>>>(acc_s, (size_t)N * DF * 2 + npad);
  k_zero<<<2048, 256, 0, stream>>>(out_h, (size_t)N * DF);

  // 2) transpose + fuse weights
  k_prep<<<128, 256, 0, stream>>>(W_text, b_text, W_inv, W_rel, b_rel,
                                  WT_text, WT_inv, WT_edge, b_fus);

  // 3) edge-side segment sums (L2-resident fp32 atomics)
  const int eb = (E + 7) / 8;                 // 8 edges per 256-thread block
  k_edge_scatter<<<eb, 256, 0, stream>>>(h, e_h, s_h, dst, acc_s, acc_r, degf, E);

  // 4) node-side WMMA GEMMs -> hsro
  const int ntiles = (N + 15) / 16;
  k_node_gemm<<<(ntiles + 1) / 2, 64, 0, stream>>>(acc_s, acc_r, degf, WT_text, b_text,
                                                   WT_inv, b_inv, hsro, N);

  // 5) h_new = segsum(hsro[src] -> dst) * norm + bias_v
  k_hnew_scatter<<<eb, 256, 0, stream>>>(hsro, src, dst, out_h, E);
  const int nb = (N + 7) / 8;
  k_hfinal<<<nb, 256, 0, stream>>>(out_h, norm, bias_v, N);

  // 6) big edge GEMM (52 GFLOP), 32 edges/wave, b64 B frags, coalesced stores
  const int etiles32 = (E + 31) / 32;
  k_edge_out<<<(etiles32 + 1) / 2, 64, 0, stream>>>(e_h, s_h, WT_edge, b_fus, out_e, E);
}